// GraphWave_17978733101641
// MI455X (gfx1250) — compile-verified
//
#include <hip/hip_runtime.h>
#include <math.h>

#define NN 20000
#define NE 200000
#define T0 13
#define SCH 256
#define ECH 512
#define HOR 12
#define EPSV 1e-5f

typedef __attribute__((ext_vector_type(2))) float v2f;
typedef __attribute__((ext_vector_type(8))) float v8f;

static __device__ __forceinline__ v8f wmma4(v2f a, v2f b, v8f c) {
  // V_WMMA_F32_16X16X4_F32 : D = A(16x4) x B(4x16) + C(16x16), all f32
  return __builtin_amdgcn_wmma_f32_16x16x4_f32(false, a, false, b, (short)0, c,
                                               false, false);
}

// ---------------- small utility kernels ----------------

__global__ void k_zero(float* __restrict__ p, size_t n) {
  size_t i = (size_t)blockIdx.x * blockDim.x + threadIdx.x;
  if (i < n) p[i] = 0.0f;
}

// permute GCN weight: dst[a*D+b] = src[m(a)*D+m(b)], m(t*C+c) = c*T+t
__global__ void k_perm_mat(const float* __restrict__ src, float* __restrict__ dst,
                           int C, int T) {
  int D = C * T;
  int idx = blockIdx.x * blockDim.x + threadIdx.x;
  if (idx >= D * D) return;
  int a = idx / D, b = idx % D;
  int ma = (a % C) * T + a / C;
  int mb = (b % C) * T + b / C;
  dst[(size_t)a * D + b] = src[(size_t)ma * D + mb];
}

__global__ void k_perm_vec(const float* __restrict__ src, float* __restrict__ dst,
                           int C, int T) {
  int D = C * T;
  int a = blockIdx.x * blockDim.x + threadIdx.x;
  if (a < D) dst[a] = src[(a % C) * T + a / C];
}

// pack filter/gate taps into Bcat[layer][k(64)][o(64)]
__global__ void k_pack_bcat(const float* __restrict__ fW,
                            const float* __restrict__ gW,
                            float* __restrict__ Bcat) {
  int idx = blockIdx.x * blockDim.x + threadIdx.x;
  if (idx >= 8 * 64 * 64) return;
  int i = idx >> 12;
  int rem = idx & 4095;
  int k = rem >> 6, o = rem & 63;
  const float* W = (o < 32) ? fW : gW;
  int oo = o & 31;
  float v = (k < 32) ? W[(((size_t)i * 32 + oo) * 32 + k) * 2 + 0]
                     : W[(((size_t)i * 32 + oo) * 32 + (k - 32)) * 2 + 1];
  Bcat[idx] = v;
}

// instance-norm stats per node over T=13
__global__ void k_stats(const float* __restrict__ x, float* __restrict__ means,
                        float* __restrict__ stdev) {
  int n = blockIdx.x * blockDim.x + threadIdx.x;
  if (n >= NN) return;
  float s = 0.f;
#pragma unroll
  for (int t = 0; t < T0; ++t) s += x[n * T0 + t];
  float m = s * (1.0f / T0);
  float s2 = 0.f;
#pragma unroll
  for (int t = 0; t < T0; ++t) {
    float d = x[n * T0 + t] - m;
    s2 += d * d;
  }
  means[n] = m;
  stdev[n] = sqrtf(s2 * (1.0f / T0) + EPSV);
}

// start 1x1 conv into h0 [N, T0, 32]
__global__ void k_start(const float* __restrict__ x, const float* __restrict__ sW,
                        const float* __restrict__ sb,
                        const float* __restrict__ means,
                        const float* __restrict__ stdev, float* __restrict__ h0) {
  int idx = blockIdx.x * blockDim.x + threadIdx.x;
  if (idx >= NN * T0 * 32) return;
  int c = idx & 31;
  int nt = idx >> 5;  // n*T0 + t
  int n = nt / T0;
  float xn = (x[nt] - means[n]) / stdev[n];
  h0[idx] = sW[c] * xn + sb[c];
}

// ---------------- ChebConv sparse part ----------------

__global__ void k_deg(const int* __restrict__ ei, const float* __restrict__ ea,
                      float* __restrict__ deg) {
  int e = blockIdx.x * blockDim.x + threadIdx.x;
  if (e >= NE) return;
  int r = ei[e], c = ei[NE + e];
  if (r != c) atomicAdd(&deg[r], ea[e]);
}

__global__ void k_dinv(const float* __restrict__ deg, float* __restrict__ dinv) {
  int n = blockIdx.x * blockDim.x + threadIdx.x;
  if (n >= NN) return;
  float d = deg[n];
  dinv[n] = (d > 0.0f) ? rsqrtf(d) : 0.0f;
}

// tx1[col,:] -= norm * xf[row,:]   (one block per edge)
__global__ void k_cheb_scatter(const int* __restrict__ ei,
                               const float* __restrict__ ea,
                               const float* __restrict__ dinv,
                               const float* __restrict__ xf,
                               float* __restrict__ tx1, int D) {
  int e = blockIdx.x;
  int row = ei[e], col = ei[NE + e];
  if (row == col) return;
  float nrm = dinv[row] * ea[e] * dinv[col];
  if (nrm == 0.0f) return;
  const float* src = xf + (size_t)row * D;
  float* dst = tx1 + (size_t)col * D;
  for (int f = threadIdx.x; f < D; f += blockDim.x)
    atomicAdd(&dst[f], -nrm * src[f]);
}

// out = X @ W0p + T1 @ W1p + bp ; each wave: 16 rows x 64 cols (4 tiles)
// grid: (ceil(1250/4), D/64), block 128
__global__ __launch_bounds__(128) void k_gemm_cheb(
    const float* __restrict__ X, const float* __restrict__ T1,
    const float* __restrict__ W0, const float* __restrict__ W1,
    const float* __restrict__ bp, float* __restrict__ out, int D) {
  int wave = threadIdx.x >> 5;
  int lane = threadIdx.x & 31;
  int l15 = lane & 15, half = lane >> 4;
  int tileM = (blockIdx.x * 4 + wave) * 16;
  if (tileM >= NN) return;  // wave-uniform, no barriers in kernel
  int colb = blockIdx.y * 64;
  const float* xrow = X + (size_t)(tileM + l15) * D;
  const float* trow = T1 + (size_t)(tileM + l15) * D;
  v8f acc0 = {}, acc1 = {}, acc2 = {}, acc3 = {};
  for (int kb = 0; kb < D; kb += 4) {
    int kk = kb + half * 2;
    const float* W0r0 = W0 + (size_t)kk * D + colb;
    const float* W0r1 = W0r0 + D;
    const float* W1r0 = W1 + (size_t)kk * D + colb;
    const float* W1r1 = W1r0 + D;
    v2f ax = *(const v2f*)(xrow + kk);
    v2f b0, b1, b2, b3;
    b0.x = W0r0[l15];      b0.y = W0r1[l15];
    b1.x = W0r0[16 + l15]; b1.y = W0r1[16 + l15];
    b2.x = W0r0[32 + l15]; b2.y = W0r1[32 + l15];
    b3.x = W0r0[48 + l15]; b3.y = W0r1[48 + l15];
    acc0 = wmma4(ax, b0, acc0);
    acc1 = wmma4(ax, b1, acc1);
    acc2 = wmma4(ax, b2, acc2);
    acc3 = wmma4(ax, b3, acc3);
    v2f at = *(const v2f*)(trow + kk);
    b0.x = W1r0[l15];      b0.y = W1r1[l15];
    b1.x = W1r0[16 + l15]; b1.y = W1r1[16 + l15];
    b2.x = W1r0[32 + l15]; b2.y = W1r1[32 + l15];
    b3.x = W1r0[48 + l15]; b3.y = W1r1[48 + l15];
    acc0 = wmma4(at, b0, acc0);
    acc1 = wmma4(at, b1, acc1);
    acc2 = wmma4(at, b2, acc2);
    acc3 = wmma4(at, b3, acc3);
  }
#pragma unroll
  for (int j = 0; j < 4; ++j) {
    int col = colb + j * 16 + l15;
    v8f acc = (j == 0) ? acc0 : (j == 1) ? acc1 : (j == 2) ? acc2 : acc3;
    float bv = bp[col];
#pragma unroll
    for (int r = 0; r < 8; ++r) {
      int row = tileM + half * 8 + r;
      out[(size_t)row * D + col] = acc[r] + bv;
    }
  }
}

// ---------------- gated dilated conv (fused WMMA GEMM) ----------------
__global__ __launch_bounds__(128) void k_conv_gated(
    const float* __restrict__ hprev, float* __restrict__ hout,
    float* __restrict__ glast, float* __restrict__ bnstats,
    const float* __restrict__ Bcat, const float* __restrict__ fb,
    const float* __restrict__ gb, int layer, int Tprev, int tout, int dil,
    int M) {
  __shared__ float ssum[32];
  __shared__ float ssq[32];
  if (threadIdx.x < 32) {
    ssum[threadIdx.x] = 0.0f;
    ssq[threadIdx.x] = 0.0f;
  }
  __syncthreads();
  int wave = threadIdx.x >> 5;
  int lane = threadIdx.x & 31;
  int l15 = lane & 15, half = lane >> 4;
  int tileM = (blockIdx.x * 4 + wave) * 16;
  float ls0 = 0.f, ls20 = 0.f, ls1 = 0.f, ls21 = 0.f;
  if (tileM < M) {
    const float* Bc = Bcat + layer * 4096;
    int rowA = tileM + l15;
    int nA = rowA / tout, tA = rowA % tout;
    const float* abase = hprev + ((size_t)nA * Tprev + tA) * 32;
    const float* abase1 = abase + (size_t)dil * 32 - 32;
    v8f acc0 = {}, acc1 = {}, acc2 = {}, acc3 = {};
    for (int kb = 0; kb < 64; kb += 4) {
      int kk = kb + half * 2;
      v2f a = (kk < 32) ? *(const v2f*)(abase + kk) : *(const v2f*)(abase1 + kk);
      const float* Br0 = Bc + kk * 64;
      const float* Br1 = Br0 + 64;
      v2f b0, b1, b2, b3;
      b0.x = Br0[l15];      b0.y = Br1[l15];
      b1.x = Br0[16 + l15]; b1.y = Br1[16 + l15];
      b2.x = Br0[32 + l15]; b2.y = Br1[32 + l15];
      b3.x = Br0[48 + l15]; b3.y = Br1[48 + l15];
      acc0 = wmma4(a, b0, acc0);
      acc1 = wmma4(a, b1, acc1);
      acc2 = wmma4(a, b2, acc2);
      acc3 = wmma4(a, b3, acc3);
    }
    int nrow[8], trow[8];
#pragma unroll
    for (int r = 0; r < 8; ++r) {
      int rowm = tileM + half * 8 + r;
      nrow[r] = rowm / tout;
      trow[r] = rowm % tout;
    }
#pragma unroll
    for (int j = 0; j < 2; ++j) {
      int c = j * 16 + l15;
      float bfv = fb[layer * 32 + c];
      float bgv = gb[layer * 32 + c];
      v8f aF = (j == 0) ? acc0 : acc1;
      v8f aG = (j == 0) ? acc2 : acc3;
#pragma unroll
      for (int r = 0; r < 8; ++r) {
        int n = nrow[r], t = trow[r];
        float f = tanhf(aF[r] + bfv);
        float g = 1.0f / (1.0f + __expf(-(aG[r] + bgv)));
        float gated = f * g;
        float res = hprev[((size_t)n * Tprev + t + dil) * 32 + c];
        float val = gated + res;
        hout[((size_t)n * tout + t) * 32 + c] = val;
        if (t == tout - 1) glast[n * 32 + c] = gated;
        if (j == 0) { ls0 += val; ls20 += val * val; }
        else        { ls1 += val; ls21 += val * val; }
      }
    }
    atomicAdd(&ssum[l15], ls0);
    atomicAdd(&ssq[l15], ls20);
    atomicAdd(&ssum[16 + l15], ls1);
    atomicAdd(&ssq[16 + l15], ls21);
  }
  __syncthreads();
  if (threadIdx.x < 32) {
    atomicAdd(&bnstats[threadIdx.x], ssum[threadIdx.x]);
    atomicAdd(&bnstats[32 + threadIdx.x], ssq[threadIdx.x]);
  }
}

__global__ void k_bn_apply(float* __restrict__ h,
                           const float* __restrict__ bnstats, float invCnt,
                           size_t total) {
  size_t i = (size_t)blockIdx.x * blockDim.x + threadIdx.x;
  if (i >= total) return;
  int c = (int)(i & 31);
  float m = bnstats[c] * invCnt;
  float v = bnstats[32 + c] * invCnt - m * m;
  h[i] = (h[i] - m) * rsqrtf(v + EPSV);
}

// skip_acc[N,256] += glast[N,32] @ skip_W[layer]^T + skip_b[layer]
// grid: (ceil(1250/4), 256/64), block 128
__global__ __launch_bounds__(128) void k_gemm_skip(
    const float* __restrict__ glast, const float* __restrict__ skipW,
    const float* __restrict__ skipb, float* __restrict__ skipacc, int layer) {
  int wave = threadIdx.x >> 5;
  int lane = threadIdx.x & 31;
  int l15 = lane & 15, half = lane >> 4;
  int tileM = (blockIdx.x * 4 + wave) * 16;
  if (tileM >= NN) return;
  int colb = blockIdx.y * 64;
  const float* A = glast + (size_t)(tileM + l15) * 32;
  const float* Wl = skipW + (size_t)layer * SCH * 32;
  v8f acc0 = {}, acc1 = {}, acc2 = {}, acc3 = {};
#pragma unroll
  for (int kb = 0; kb < 32; kb += 4) {
    int kk = kb + half * 2;
    v2f a = *(const v2f*)(A + kk);
    // W[o][k] contiguous in k -> one v2f per tile
    v2f b0 = *(const v2f*)(Wl + (size_t)(colb + l15) * 32 + kk);
    v2f b1 = *(const v2f*)(Wl + (size_t)(colb + 16 + l15) * 32 + kk);
    v2f b2 = *(const v2f*)(Wl + (size_t)(colb + 32 + l15) * 32 + kk);
    v2f b3 = *(const v2f*)(Wl + (size_t)(colb + 48 + l15) * 32 + kk);
    acc0 = wmma4(a, b0, acc0);
    acc1 = wmma4(a, b1, acc1);
    acc2 = wmma4(a, b2, acc2);
    acc3 = wmma4(a, b3, acc3);
  }
#pragma unroll
  for (int j = 0; j < 4; ++j) {
    int col = colb + j * 16 + l15;
    v8f acc = (j == 0) ? acc0 : (j == 1) ? acc1 : (j == 2) ? acc2 : acc3;
    float bv = skipb[layer * SCH + col];
#pragma unroll
    for (int r = 0; r < 8; ++r) {
      int row = tileM + half * 8 + r;
      size_t o = (size_t)row * SCH + col;
      skipacc[o] += acc[r] + bv;
    }
  }
}

// out1[N,512] = relu(skip)[N,256] @ end1_W^T + end1_b
// grid: (ceil(1250/4), 512/64), block 128
__global__ __launch_bounds__(128) void k_gemm_end1(
    const float* __restrict__ skip, const float* __restrict__ e1W,
    const float* __restrict__ e1b, float* __restrict__ out1) {
  int wave = threadIdx.x >> 5;
  int lane = threadIdx.x & 31;
  int l15 = lane & 15, half = lane >> 4;
  int tileM = (blockIdx.x * 4 + wave) * 16;
  if (tileM >= NN) return;
  int colb = blockIdx.y * 64;
  const float* A = skip + (size_t)(tileM + l15) * SCH;
  v8f acc0 = {}, acc1 = {}, acc2 = {}, acc3 = {};
  for (int kb = 0; kb < SCH; kb += 4) {
    int kk = kb + half * 2;
    v2f a = *(const v2f*)(A + kk);
    a.x = fmaxf(a.x, 0.0f);
    a.y = fmaxf(a.y, 0.0f);
    // end1_W is [512,256], contiguous in k
    v2f b0 = *(const v2f*)(e1W + (size_t)(colb + l15) * SCH + kk);
    v2f b1 = *(const v2f*)(e1W + (size_t)(colb + 16 + l15) * SCH + kk);
    v2f b2 = *(const v2f*)(e1W + (size_t)(colb + 32 + l15) * SCH + kk);
    v2f b3 = *(const v2f*)(e1W + (size_t)(colb + 48 + l15) * SCH + kk);
    acc0 = wmma4(a, b0, acc0);
    acc1 = wmma4(a, b1, acc1);
    acc2 = wmma4(a, b2, acc2);
    acc3 = wmma4(a, b3, acc3);
  }
#pragma unroll
  for (int j = 0; j < 4; ++j) {
    int col = colb + j * 16 + l15;
    v8f acc = (j == 0) ? acc0 : (j == 1) ? acc1 : (j == 2) ? acc2 : acc3;
    float bv = e1b[col];
#pragma unroll
    for (int r = 0; r < 8; ++r) {
      int row = tileM + half * 8 + r;
      out1[(size_t)row * ECH + col] = acc[r] + bv;
    }
  }
}

// final 512 -> 12 projection + de-normalization
__global__ void k_end2(const float* __restrict__ out1, const float* __restrict__ e2W,
                       const float* __restrict__ e2b,
                       const float* __restrict__ stdev,
                       const float* __restrict__ means, float* __restrict__ out) {
  int idx = blockIdx.x * blockDim.x + threadIdx.x;
  if (idx >= NN * HOR) return;
  int n = idx / HOR, h = idx % HOR;
  const float* r = out1 + (size_t)n * ECH;
  const float* w = e2W + (size_t)h * ECH;
  float s = e2b[h];
#pragma unroll 8
  for (int k = 0; k < ECH; ++k) s += r[k] * w[k];
  out[idx] = s * stdev[n] + means[n];
}

// ---------------- host launch ----------------

extern "C" void kernel_launch(void* const* d_in, const int* in_sizes, int n_in,
                              void* d_out, int out_size, void* d_ws,
                              size_t ws_size, hipStream_t stream) {
  (void)in_sizes; (void)n_in; (void)out_size; (void)ws_size;
  const float* x = (const float*)d_in[0];
  const int* ei = (const int*)d_in[1];
  const float* ea = (const float*)d_in[2];
  const float* start_W = (const float*)d_in[3];
  const float* start_b = (const float*)d_in[4];
  const float* filter_W = (const float*)d_in[5];
  const float* filter_b = (const float*)d_in[6];
  const float* gate_W = (const float*)d_in[7];
  const float* gate_b = (const float*)d_in[8];
  const float* skip_W = (const float*)d_in[9];
  const float* skip_b = (const float*)d_in[10];
  const float* g0W0 = (const float*)d_in[11];
  const float* g0W1 = (const float*)d_in[12];
  const float* g0b = (const float*)d_in[13];
  const float* g1W0 = (const float*)d_in[14];
  const float* g1W1 = (const float*)d_in[15];
  const float* g1b = (const float*)d_in[16];
  const float* e1W = (const float*)d_in[17];
  const float* e1b = (const float*)d_in[18];
  const float* e2W = (const float*)d_in[19];
  const float* e2b = (const float*)d_in[20];
  float* out = (float*)d_out;

  // workspace carve
  char* base = (char*)d_ws;
  size_t off = 0;
  auto alloc = [&](size_t bytes) -> float* {
    float* p = (float*)(base + off);
    off += (bytes + 255) & ~(size_t)255;
    return p;
  };
  float* h_a = alloc((size_t)NN * T0 * 32 * 4);
  float* h_b = alloc((size_t)NN * T0 * 32 * 4);
  float* big = alloc((size_t)NN * ECH * 4);  // tx1 (<=N*384) then out1 (N*512)
  float* skip_acc = alloc((size_t)NN * SCH * 4);
  float* glast = alloc((size_t)NN * 32 * 4);
  float* meansb = alloc((size_t)NN * 4);
  float* stdevb = alloc((size_t)NN * 4);
  float* degb = alloc((size_t)NN * 4);
  float* dinvb = alloc((size_t)NN * 4);
  float* bnstats = alloc(64 * 4);
  float* W0p0 = alloc(384 * 384 * 4);
  float* W1p0 = alloc(384 * 384 * 4);
  float* bp0 = alloc(384 * 4);
  float* W0p1 = alloc(192 * 192 * 4);
  float* W1p1 = alloc(192 * 192 * 4);
  float* bp1 = alloc(192 * 4);
  float* Bcat = alloc(8 * 64 * 64 * 4);

  auto zero = [&](float* p, size_t n) {
    k_zero<<<(unsigned)((n + 255) / 256), 256, 0, stream>>>(p, n);
  };

  // prep: permuted GCN weights, packed conv taps, graph degree
  k_perm_mat<<<(384 * 384 + 255) / 256, 256, 0, stream>>>(g0W0, W0p0, 32, 12);
  k_perm_mat<<<(384 * 384 + 255) / 256, 256, 0, stream>>>(g0W1, W1p0, 32, 12);
  k_perm_vec<<<2, 256, 0, stream>>>(g0b, bp0, 32, 12);
  k_perm_mat<<<(192 * 192 + 255) / 256, 256, 0, stream>>>(g1W0, W0p1, 32, 6);
  k_perm_mat<<<(192 * 192 + 255) / 256, 256, 0, stream>>>(g1W1, W1p1, 32, 6);
  k_perm_vec<<<1, 256, 0, stream>>>(g1b, bp1, 32, 6);
  k_pack_bcat<<<(8 * 64 * 64 + 255) / 256, 256, 0, stream>>>(filter_W, gate_W, Bcat);
  zero(degb, NN);
  k_deg<<<(NE + 255) / 256, 256, 0, stream>>>(ei, ea, degb);
  k_dinv<<<(NN + 255) / 256, 256, 0, stream>>>(degb, dinvb);
  zero(skip_acc, (size_t)NN * SCH);

  // instance norm + start conv
  k_stats<<<(NN + 255) / 256, 256, 0, stream>>>(x, meansb, stdevb);
  k_start<<<(NN * T0 * 32 + 255) / 256, 256, 0, stream>>>(x, start_W, start_b,
                                                          meansb, stdevb, h_a);

  const int DIL[8] = {1, 2, 1, 2, 1, 2, 1, 2};
  float* cur = h_a;
  float* alt = h_b;
  int T = T0;
  int mBlocks = (NN / 16 + 3) / 4;  // 1250 M-tiles, 4 per block
  for (int i = 0; i < 8; ++i) {
    if (i == 1 || i == 5) {
      int D = 32 * T;  // 384 or 192
      const float* W0p = (i == 1) ? W0p0 : W0p1;
      const float* W1p = (i == 1) ? W1p0 : W1p1;
      const float* bp = (i == 1) ? bp0 : bp1;
      float* tx1 = big;
      zero(tx1, (size_t)NN * D);
      k_cheb_scatter<<<NE, 128, 0, stream>>>(ei, ea, dinvb, cur, tx1, D);
      dim3 g(mBlocks, D / 64);
      k_gemm_cheb<<<g, 128, 0, stream>>>(cur, tx1, W0p, W1p, bp, alt, D);
      float* t = cur; cur = alt; alt = t;
    }
    int d = DIL[i];
    int tout = T - d;
    int M = NN * tout;
    zero(bnstats, 64);
    k_conv_gated<<<(M + 63) / 64, 128, 0, stream>>>(
        cur, alt, glast, bnstats, Bcat, filter_b, gate_b, i, T, tout, d, M);
    dim3 gs(mBlocks, SCH / 64);
    k_gemm_skip<<<gs, 128, 0, stream>>>(glast, skip_W, skip_b, skip_acc, i);
    size_t tot = (size_t)M * 32;
    k_bn_apply<<<(unsigned)((tot + 255) / 256), 256, 0, stream>>>(
        alt, bnstats, 1.0f / (float)M, tot);
    float* t = cur; cur = alt; alt = t;
    T = tout;
  }

  // end stage
  float* out1 = big;  // GCN scratch no longer needed
  dim3 ge(mBlocks, ECH / 64);
  k_gemm_end1<<<ge, 128, 0, stream>>>(skip_acc, e1W, e1b, out1);
  k_end2<<<(NN * HOR + 255) / 256, 256, 0, stream>>>(out1, e2W, e2b, stdevb,
                                                     meansb, out);
}